// ScaleDotProductAttention_31516470018404
// MI455X (gfx1250) — compile-verified
//
#include <hip/hip_runtime.h>
#include <math.h>

// Scaled dot-product attention (fwd, returns out AND full attn matrix)
// B=2, H=16, S=2048, D=64, fp32.
// One wave per 16-row query tile (single-wave workgroups => barriers are NOPs).
// Two-pass online softmax with score recompute so attn memory is written
// exactly once, staged through LDS and flushed as global_store_b128.
// All matmuls via V_WMMA_F32_16X16X4_F32 (full fp32 precision).

typedef __attribute__((ext_vector_type(2))) float v2f;
typedef __attribute__((ext_vector_type(4))) float v4f;
typedef __attribute__((ext_vector_type(8))) float v8f;

#define S_LEN   2048
#define D_DIM   64
#define KTILES  (S_LEN / 16)
#define GROUP   4        // k-tiles staged in LDS between attn flushes
#define LDS_W   68       // row stride (floats): 16B-aligned rows, conflict-free banks

__global__ __launch_bounds__(32)
void sdpa_wmma_f32_kernel(const float* __restrict__ q,
                          const float* __restrict__ k,
                          const float* __restrict__ v,
                          const int*   __restrict__ mask,
                          float* __restrict__ out,
                          float* __restrict__ attn)
{
    __shared__ float pStage[16 * LDS_W];   // 16 x 64 staged tile (padded rows)

    const int lane = threadIdx.x & 31;
    const int half = lane >> 4;            // 0: lanes 0-15, 1: lanes 16-31
    const int l16  = lane & 15;

    const int bh    = blockIdx.x / KTILES; // 0..B*H-1
    const int qtile = blockIdx.x % KTILES;
    const int b     = bh >> 4;             // H == 16
    const int qbase = qtile * 16;

    const float scale = 0.125f;            // 1/sqrt(64)

    const float* qBH = q + (size_t)bh * S_LEN * D_DIM;
    const float* kBH = k + (size_t)bh * S_LEN * D_DIM;
    const float* vBH = v + (size_t)bh * S_LEN * D_DIM;
    const int*   mB  = mask + (size_t)b * S_LEN * S_LEN;
    float* outBH  = out  + (size_t)bh * S_LEN * D_DIM;
    float* attnBH = attn + (size_t)bh * S_LEN * S_LEN;

    // ---- Q tile as sixteen 16x4 A-fragments, resident in VGPRs for both passes.
    //      A-layout: lane L holds A[L%16][4*t + 2*(L/16) + j], j=0,1.
    v2f qa[16];
    {
        const float* qrow = qBH + (size_t)(qbase + l16) * D_DIM + 2 * half;
        #pragma unroll
        for (int t = 0; t < 16; ++t)
            qa[t] = *(const v2f*)(qrow + 4 * t);
    }

    // Per-LANE online softmax state (row = r + 8*half in C layout); the
    // cross-lane merge happens once after pass 1, not per tile.
    float m_ln[8], l_ln[8];
    #pragma unroll
    for (int r = 0; r < 8; ++r) { m_ln[r] = -3.0e38f; l_ln[r] = 0.0f; }

    // ================= Pass 1: per-lane online row max / sum(exp) ============
    for (int kt = 0; kt < KTILES; ++kt) {
        const int kbase = kt * 16;
        const float* krow = kBH + (size_t)(kbase + l16) * D_DIM + 2 * half;
        v8f c = {};
        #pragma unroll
        for (int t = 0; t < 16; ++t) {
            v2f kb = *(const v2f*)(krow + 4 * t);
            c = __builtin_amdgcn_wmma_f32_16x16x4_f32(
                    false, qa[t], false, kb, (short)0, c, false, false);
        }
        #pragma unroll
        for (int r = 0; r < 8; ++r) {
            const int qrow = qbase + r + 8 * half;
            const int mv = mB[(size_t)qrow * S_LEN + kbase + l16];
            const float s = (mv == 0) ? -10000.0f : c[r] * scale;
            const float mo = m_ln[r];
            const float mn = fmaxf(mo, s);
            l_ln[r] = l_ln[r] * __expf(mo - mn) + __expf(s - mn);
            m_ln[r] = mn;
        }
    }

    // ---- Cross-lane merge (once): true row max + rescaled sum over 16 lanes.
    float m_row[8], inv_l[8];
    #pragma unroll
    for (int r = 0; r < 8; ++r) {
        float m = m_ln[r];
        #pragma unroll
        for (int off = 1; off < 16; off <<= 1)
            m = fmaxf(m, __shfl_xor(m, off, 32));
        float lc = l_ln[r] * __expf(m_ln[r] - m);
        #pragma unroll
        for (int off = 1; off < 16; off <<= 1)
            lc += __shfl_xor(lc, off, 32);
        m_row[r] = m;
        inv_l[r] = 1.0f / lc;
    }

    // Output accumulators: out[16 x 64] = four 16x16 C tiles.
    v8f oacc[4];
    #pragma unroll
    for (int n = 0; n < 4; ++n) oacc[n] = (v8f){};

    // ====== Pass 2: recompute scores, stage probs in LDS, P@V via WMMA,
    //        flush attn as wide b128 stores every GROUP k-tiles. =============
    for (int kg = 0; kg < KTILES; kg += GROUP) {
        #pragma unroll
        for (int ks = 0; ks < GROUP; ++ks) {
            const int kbase = (kg + ks) * 16;
            const float* krow = kBH + (size_t)(kbase + l16) * D_DIM + 2 * half;
            v8f c = {};
            #pragma unroll
            for (int t = 0; t < 16; ++t) {
                v2f kb = *(const v2f*)(krow + 4 * t);
                c = __builtin_amdgcn_wmma_f32_16x16x4_f32(
                        false, qa[t], false, kb, (short)0, c, false, false);
            }
            // Normalized probabilities into this tile's LDS slot.
            #pragma unroll
            for (int r = 0; r < 8; ++r) {
                const int qrow = qbase + r + 8 * half;
                const int mv = mB[(size_t)qrow * S_LEN + kbase + l16];
                const float s = (mv == 0) ? -10000.0f : c[r] * scale;
                const float p = __expf(s - m_row[r]) * inv_l[r];
                pStage[(r + 8 * half) * LDS_W + ks * 16 + l16] = p;
            }
            __syncthreads();   // single-wave WG: S_NOP; orders DS store->load
            // oacc[n] += P(16x16) * V(16x64): C-layout -> A-layout via LDS.
            #pragma unroll
            for (int tp = 0; tp < 4; ++tp) {
                v2f pa = *(const v2f*)&pStage[l16 * LDS_W + ks * 16 + 4 * tp + 2 * half];
                #pragma unroll
                for (int n = 0; n < 4; ++n) {
                    const float* vptr = vBH + (size_t)(kbase + 4 * tp + 2 * half) * D_DIM
                                            + n * 16 + l16;
                    v2f vb;
                    vb.x = vptr[0];
                    vb.y = vptr[D_DIM];
                    oacc[n] = __builtin_amdgcn_wmma_f32_16x16x4_f32(
                                  false, pa, false, vb, (short)0, oacc[n], false, false);
                }
            }
        }
        __syncthreads();
        // Flush staged 16x64 prob block: each half-wave stores a contiguous
        // 256B row segment (global_store_b128).
        #pragma unroll
        for (int i = 0; i < 8; ++i) {
            const int idx4 = i * 32 + lane;      // 0..255 float4 chunks
            const int row  = idx4 >> 4;          // 16 float4 per row
            const int col  = (idx4 & 15) * 4;
            v4f pv = *(const v4f*)&pStage[row * LDS_W + col];
            *(v4f*)&attnBH[(size_t)(qbase + row) * S_LEN + kg * 16 + col] = pv;
        }
        __syncthreads();   // staging reads done before next group overwrites
    }

    // ---- Out tile: restage through LDS and store as b128 as well.
    #pragma unroll
    for (int n = 0; n < 4; ++n)
        #pragma unroll
        for (int r = 0; r < 8; ++r)
            pStage[(r + 8 * half) * LDS_W + n * 16 + l16] = oacc[n][r];
    __syncthreads();
    #pragma unroll
    for (int i = 0; i < 8; ++i) {
        const int idx4 = i * 32 + lane;
        const int row  = idx4 >> 4;
        const int col  = (idx4 & 15) * 4;
        v4f ov = *(const v4f*)&pStage[row * LDS_W + col];
        *(v4f*)&outBH[(size_t)(qbase + row) * D_DIM + col] = ov;
    }
}

extern "C" void kernel_launch(void* const* d_in, const int* in_sizes, int n_in,
                              void* d_out, int out_size, void* d_ws, size_t ws_size,
                              hipStream_t stream) {
    (void)in_sizes; (void)n_in; (void)d_ws; (void)ws_size; (void)out_size;
    const float* q    = (const float*)d_in[0];
    const float* k    = (const float*)d_in[1];
    const float* v    = (const float*)d_in[2];
    const int*   mask = (const int*)d_in[3];

    const int B = 2, H = 16;
    float* out  = (float*)d_out;                         // [B,H,S,D]
    float* attn = out + (size_t)B * H * S_LEN * D_DIM;   // [B,H,S,S]

    const int blocks = B * H * KTILES;                   // one wave per q-tile
    sdpa_wmma_f32_kernel<<<blocks, 32, 0, stream>>>(q, k, v, mask, out, attn);
}